// CFChurn_si_only_89859305767633
// MI455X (gfx1250) — compile-verified
//
#include <hip/hip_runtime.h>
#include <hip/hip_bf16.h>

// Problem constants (from reference)
static constexpr int kN  = 100000;   // nodes
static constexpr int kE  = 1600000;  // edges
static constexpr int kNT = 80000;    // treated rows
static constexpr int kFD = 54;
static constexpr int kNH = 48;       // FD - 6

typedef float v2f __attribute__((ext_vector_type(2)));
typedef float v8f __attribute__((ext_vector_type(8)));

__device__ __forceinline__ float lrelu(float v) { return v >= 0.0f ? v : 0.01f * v; }

// ---------------- embedding gather: X[:, 0:48] = emb_tables[j, discrete_x[i,j]] ----
__global__ void k_gather(const int* __restrict__ dx, const float* __restrict__ emb,
                         float* __restrict__ X /* ld 96 */)
{
    size_t idx = (size_t)blockIdx.x * blockDim.x + threadIdx.x;
    if (idx >= (size_t)kN * kNH) return;
    int i = (int)(idx / kNH), j = (int)(idx % kNH);
    int v = dx[(size_t)i * kFD + j];
    X[(size_t)i * 96 + j] = emb[j * 100 + v];
}

// ---------------- degree / rsqrt ----------------
__global__ void k_deg(const int* __restrict__ dst, float* __restrict__ deg)
{
    int e = blockIdx.x * blockDim.x + threadIdx.x;
    if (e < kE) atomicAdd(&deg[dst[e]], 1.0f);
}

__global__ void k_di(const float* __restrict__ deg, float* __restrict__ di)
{
    int i = blockIdx.x * blockDim.x + threadIdx.x;
    if (i < kN) di[i] = rsqrtf(deg[i] + 1.0f);
}

// ------- edge scatter: agg[dst] += xw[src]*di[src]*di[dst]; 4 features/thread -----
__global__ void k_scatter(const int* __restrict__ src, const int* __restrict__ dst,
                          const float* __restrict__ di, const float4* __restrict__ xw4,
                          float* __restrict__ agg)
{
    size_t t = (size_t)blockIdx.x * blockDim.x + threadIdx.x;
    if (t >= (size_t)kE * 12) return;                 // 48 feats = 12 float4 per edge
    int e = (int)(t / 12), q = (int)(t % 12);
    int s = src[e], d = dst[e];
    float c = di[s] * di[d];
    float4 v = xw4[(size_t)s * 12 + q];
    float* ap = agg + (size_t)d * kNH + q * 4;
    atomicAdd(ap + 0, v.x * c);
    atomicAdd(ap + 1, v.y * c);
    atomicAdd(ap + 2, v.z * c);
    atomicAdd(ap + 3, v.w * c);
}

// ---------------- gcn combine: Xg = lrelu(agg + xw/deg + b) ----------------------
__global__ void k_combine(const float* __restrict__ xw, const float* __restrict__ agg,
                          const float* __restrict__ di, const float* __restrict__ b,
                          float* __restrict__ Xg /* X+48, ld 96 */)
{
    size_t idx = (size_t)blockIdx.x * blockDim.x + threadIdx.x;
    if (idx >= (size_t)kN * kNH) return;
    int i = (int)(idx / kNH), j = (int)(idx % kNH);
    float d = di[i];
    Xg[(size_t)i * 96 + j] = lrelu(agg[idx] + xw[idx] * d * d + b[j]);
}

// ---------------- generic fp32 WMMA GEMM: C = act(A@W + b) [+ R] ------------------
// FLAGS bit0: leaky-relu, bit1: residual add (after activation, per reference).
// K, NC compile-time -> full unroll, constant LDS offsets.
// Block = 256 threads = 8 waves; wave owns a 16-row slab and ALL NC/16 column tiles
// (accumulators in registers), so each A fragment is loaded from global exactly once.
// W repacked in LDS as K-pairs: sW[(pair*NC + n)*2 + parity] -> one ds_load_b64/frag.
template<int FLAGS, int K, int NC>
__global__ void k_gemm(const float* __restrict__ A, int lda,
                       const float* __restrict__ W,
                       const float* __restrict__ bias,
                       float* __restrict__ C, int ldc,
                       const float* __restrict__ R, int ldr,
                       int M)
{
    constexpr int NT16 = NC / 16;
    extern __shared__ float sW[]; // K*NC floats, pair-packed
    for (int i = threadIdx.x; i < K * NC; i += blockDim.x) {
        int k = i / NC, n = i % NC;
        sW[((k >> 1) * NC + n) * 2 + (k & 1)] = W[i];
    }
    __syncthreads();

    const int wave = threadIdx.x >> 5;
    const int lane = threadIdx.x & 31;
    const int half = lane >> 4;   // 0: lanes 0-15 (K=k,k+1), 1: lanes 16-31 (K=k+2,k+3)
    const int l16  = lane & 15;
    const int row0 = blockIdx.x * 128 + wave * 16;

    int arow = row0 + l16;
    if (arow >= M) arow = M - 1;               // clamp loads; stores masked below
    const float* Ar = A + (size_t)arow * lda + 2 * half;

    v8f acc[NT16];
    #pragma unroll
    for (int c = 0; c < NT16; ++c) {
        float bv = bias ? bias[c * 16 + l16] : 0.0f;
        #pragma unroll
        for (int r = 0; r < 8; ++r) acc[c][r] = bv;
    }

    #pragma unroll
    for (int k = 0; k < K; k += 4) {
        v2f a = *(const v2f*)(Ar + k);         // 8B global load, per-lane A fragment
        const float* bp = sW + ((k >> 1) + half) * (NC * 2) + l16 * 2;
        #pragma unroll
        for (int c = 0; c < NT16; ++c) {
            v2f b = *(const v2f*)(bp + c * 32); // ds_load_b64, imm offset
            acc[c] = __builtin_amdgcn_wmma_f32_16x16x4_f32(false, a, false, b,
                                                           (short)0, acc[c], false, false);
        }
    }

    // C/D layout: VGPR r, lane L -> row = r + 8*(L>=16), col = L%16
    #pragma unroll
    for (int c = 0; c < NT16; ++c) {
        int n = c * 16 + l16;
        #pragma unroll
        for (int r = 0; r < 8; ++r) {
            int m = row0 + half * 8 + r;
            if (m < M) {
                float v = acc[c][r];
                if (FLAGS & 1) v = lrelu(v);
                if (FLAGS & 2) v += R[(size_t)m * ldr + n];
                C[(size_t)m * ldc + n] = v;
            }
        }
    }
}

// ---------------- cross layers: one wave32 per row (96 = 3 floats/lane) -----------
__global__ void k_cross(const float* __restrict__ X, const float* __restrict__ w1,
                        const float* __restrict__ cb1, const float* __restrict__ w2,
                        const float* __restrict__ cb2, float* __restrict__ OUT /* HC+96, ld192 */)
{
    int gw   = (int)(((size_t)blockIdx.x * blockDim.x + threadIdx.x) >> 5);
    int lane = threadIdx.x & 31;
    if (gw >= kN) return;
    const float* xr = X + (size_t)gw * 96;
    float x0 = xr[lane], x1 = xr[lane + 32], x2 = xr[lane + 64];

    float p = x0 * w1[lane] + x1 * w1[lane + 32] + x2 * w1[lane + 64];
    #pragma unroll
    for (int o = 16; o > 0; o >>= 1) p += __shfl_xor(p, o, 32);

    float l0 = x0 * p + cb1[lane]      + x0;
    float l1 = x1 * p + cb1[lane + 32] + x1;
    float l2 = x2 * p + cb1[lane + 64] + x2;

    float q = l0 * w2[lane] + l1 * w2[lane + 32] + l2 * w2[lane + 64];
    #pragma unroll
    for (int o = 16; o > 0; o >>= 1) q += __shfl_xor(q, o, 32);

    l0 = x0 * q + cb2[lane]      + l0;
    l1 = x1 * q + cb2[lane + 32] + l1;
    l2 = x2 * q + cb2[lane + 64] + l2;

    float* orow = OUT + (size_t)gw * 192;
    orow[lane] = l0; orow[lane + 32] = l1; orow[lane + 64] = l2;
}

// ---------------- final heads: pred_T, y0/y1 projections + treatment mix ----------
__global__ void k_final(const float* __restrict__ hsi, const float* __restrict__ z0,
                        const float* __restrict__ z1,
                        const float* __restrict__ TW,  const float* __restrict__ Tb,
                        const float* __restrict__ W03, const float* __restrict__ b03,
                        const float* __restrict__ W13, const float* __restrict__ b13,
                        const int* __restrict__ t, float* __restrict__ out)
{
    int r    = (int)(((size_t)blockIdx.x * blockDim.x + threadIdx.x) >> 5);
    int lane = threadIdx.x & 31;
    if (r >= kNT) return;
    const float* h = hsi + (size_t)r * 64;
    float pt = h[lane] * TW[lane] + h[lane + 32] * TW[lane + 32];
    float a0 = z0[(size_t)r * 64 + lane] * W03[lane] + z0[(size_t)r * 64 + lane + 32] * W03[lane + 32];
    float a1 = z1[(size_t)r * 64 + lane] * W13[lane] + z1[(size_t)r * 64 + lane + 32] * W13[lane + 32];
    #pragma unroll
    for (int o = 16; o > 0; o >>= 1) {
        pt += __shfl_xor(pt, o, 32);
        a0 += __shfl_xor(a0, o, 32);
        a1 += __shfl_xor(a1, o, 32);
    }
    if (lane == 0) {
        pt += Tb[0]; a0 += b03[0]; a1 += b13[0];
        float tf = (float)t[r];
        out[r]           = (1.0f - tf) * a0 + tf * a1;          // y_f
        out[kNT + r]     = tf * a0 + (1.0f - tf) * a1;          // y_cf
        out[2 * kNT + r] = pt;                                  // pred_T
    }
}

#define GEMM(F, Kc, Ncc, A, lda, W, b, C, ldc, R, ldr, M) \
    k_gemm<F, Kc, Ncc><<<((M) + 127) / 128, 256, (size_t)(Kc) * (Ncc) * sizeof(float), stream>>>( \
        A, lda, W, b, C, ldc, R, ldr, M)

extern "C" void kernel_launch(void* const* d_in, const int* in_sizes, int n_in,
                              void* d_out, int out_size, void* d_ws, size_t ws_size,
                              hipStream_t stream)
{
    (void)in_sizes; (void)n_in; (void)out_size; (void)ws_size;

    const int*   dx   = (const int*)  d_in[0];
    const int*   ei   = (const int*)  d_in[2];
    const int*   tarr = (const int*)  d_in[5];
    const float* emb  = (const float*)d_in[6];
    const float* g0W  = (const float*)d_in[7];  const float* g0b = (const float*)d_in[8];
    const float* g1W  = (const float*)d_in[9];  const float* g1b = (const float*)d_in[10];
    const float* g2W  = (const float*)d_in[11]; const float* g2b = (const float*)d_in[12];
    const float* rW1  = (const float*)d_in[13]; const float* rb1 = (const float*)d_in[14];
    const float* rW2  = (const float*)d_in[15]; const float* rb2 = (const float*)d_in[16];
    const float* cw1  = (const float*)d_in[17]; const float* cb1 = (const float*)d_in[18];
    const float* cw2  = (const float*)d_in[19]; const float* cb2 = (const float*)d_in[20];
    const float* f1W  = (const float*)d_in[21]; const float* f1b = (const float*)d_in[22];
    const float* f2W  = (const float*)d_in[23]; const float* f2b = (const float*)d_in[24];
    const float* y0W1 = (const float*)d_in[25]; const float* y0b1 = (const float*)d_in[26];
    const float* y0W2 = (const float*)d_in[27]; const float* y0b2 = (const float*)d_in[28];
    const float* y0W3 = (const float*)d_in[29]; const float* y0b3 = (const float*)d_in[30];
    const float* y1W1 = (const float*)d_in[31]; const float* y1b1 = (const float*)d_in[32];
    const float* y1W2 = (const float*)d_in[33]; const float* y1b2 = (const float*)d_in[34];
    const float* y1W3 = (const float*)d_in[35]; const float* y1b3 = (const float*)d_in[36];
    const float* TW   = (const float*)d_in[37]; const float* Tb   = (const float*)d_in[38];

    float* out = (float*)d_out;
    float* ws  = (float*)d_ws;

    // Workspace layout (floats), peak ~155 MB with reuse:
    float* X   = ws;                           // N*96  : [x_d | x_g]; later reused as h1
    float* XW  = ws + (size_t)kN * 96;         // N*48  : GCN xw
    float* AGG = XW + (size_t)kN * 48;         // N*48  : GCN agg      (XW|AGG -> H, N*96)
    float* HC  = ws + (size_t)kN * 192;        // N*192 : [h | xl]; later reused for heads
    float* DEG = ws + (size_t)kN * 384;        // N
    float* DI  = DEG + kN;                     // N
    float* H   = XW;                           // N*96 (alias, after GCN done)
    float* H1  = X;                            // N*96 (alias, after cross done)
    float* ZA  = HC;                           // NT*64 (alias, after h_si done)
    float* ZB0 = HC + (size_t)kNT * 64;        // NT*64
    float* ZB1 = ZB0 + (size_t)kNT * 64;       // NT*64
    float* HSI = out + 3 * kNT;                // N*64 lives directly in d_out

    const int* src = ei;
    const int* dst = ei + kE;

    // --- embedding + degree ---
    hipMemsetAsync(DEG, 0, (size_t)kN * sizeof(float), stream);
    k_gather<<<(kN * kNH + 255) / 256, 256, 0, stream>>>(dx, emb, X);
    k_deg<<<(kE + 255) / 256, 256, 0, stream>>>(dst, DEG);
    k_di<<<(kN + 255) / 256, 256, 0, stream>>>(DEG, DI);

    // --- g0: x_g = lrelu(x_d @ g0_W + b), written into X[:,48:96] ---
    GEMM(1, 48, 48, X, 96, g0W, g0b, X + 48, 96, nullptr, 0, kN);

    // --- GCN layer 1 ---
    GEMM(0, 48, 48, X + 48, 96, g1W, nullptr, XW, 48, nullptr, 0, kN);
    hipMemsetAsync(AGG, 0, (size_t)kN * 48 * sizeof(float), stream);
    k_scatter<<<(int)(((size_t)kE * 12 + 255) / 256), 256, 0, stream>>>(
        src, dst, DI, (const float4*)XW, AGG);
    k_combine<<<(kN * kNH + 255) / 256, 256, 0, stream>>>(XW, AGG, DI, g1b, X + 48);

    // --- GCN layer 2 ---
    GEMM(0, 48, 48, X + 48, 96, g2W, nullptr, XW, 48, nullptr, 0, kN);
    hipMemsetAsync(AGG, 0, (size_t)kN * 48 * sizeof(float), stream);
    k_scatter<<<(int)(((size_t)kE * 12 + 255) / 256), 256, 0, stream>>>(
        src, dst, DI, (const float4*)XW, AGG);
    k_combine<<<(kN * kNH + 255) / 256, 256, 0, stream>>>(XW, AGG, DI, g2b, X + 48);

    // --- residual MLP: h = lrelu(xW+b)+x (x2), second write into HC[:, :96] ---
    GEMM(3, 96, 96, X, 96, rW1, rb1, H, 96, X, 96, kN);
    GEMM(3, 96, 96, H, 96, rW2, rb2, HC, 192, H, 96, kN);

    // --- cross layers into HC[:, 96:192] ---
    k_cross<<<(int)(((size_t)kN * 32 + 255) / 256), 256, 0, stream>>>(X, cw1, cb1, cw2, cb2, HC + 96);

    // --- fusion: h1 = lrelu(HC @ f1_W + b); h_si = lrelu(h1 @ f2_W + b) -> d_out ---
    GEMM(1, 192, 96, HC, 192, f1W, f1b, H1, 96, nullptr, 0, kN);
    GEMM(1, 96, 64, H1, 96, f2W, f2b, HSI, 64, nullptr, 0, kN);

    // --- outcome heads (only first NT rows needed) ---
    GEMM(1, 64, 64, HSI, 64, y0W1, y0b1, ZA,  64, nullptr, 0, kNT);
    GEMM(1, 64, 64, ZA,  64, y0W2, y0b2, ZB0, 64, nullptr, 0, kNT);
    GEMM(1, 64, 64, HSI, 64, y1W1, y1b1, ZA,  64, nullptr, 0, kNT);
    GEMM(1, 64, 64, ZA,  64, y1W2, y1b2, ZB1, 64, nullptr, 0, kNT);

    // --- final projections + treatment mix ---
    k_final<<<(int)(((size_t)kNT * 32 + 255) / 256), 256, 0, stream>>>(
        HSI, ZB0, ZB1, TW, Tb, y0W3, y0b3, y1W3, y1b3, tarr, out);
}